// Renderer_74732430950751
// MI455X (gfx1250) — compile-verified
//
#include <hip/hip_runtime.h>
#include <hip/hip_bf16.h>

#define NT 256   // threads per block (8 wave32s)
#define TV 32    // vertices per block
#define NC 6     // faces blended per vertex

// CDNA5 async global->LDS copies (ASYNCcnt path). VDST = LDS byte-offset VGPR,
// VADDR = 64-bit global address VGPR pair, no SADDR.
__device__ __forceinline__ void async_g2l_b32(void* lds_dst, const void* gsrc) {
  unsigned loff =
      (unsigned)(unsigned long)(__attribute__((address_space(3))) void*)lds_dst;
  unsigned long long gaddr = (unsigned long long)gsrc;
  asm volatile("global_load_async_to_lds_b32 %0, %1, off"
               :: "v"(loff), "v"(gaddr) : "memory");
}

__device__ __forceinline__ void async_g2l_b128(void* lds_dst, const void* gsrc) {
  unsigned loff =
      (unsigned)(unsigned long)(__attribute__((address_space(3))) void*)lds_dst;
  unsigned long long gaddr = (unsigned long long)gsrc;
  asm volatile("global_load_async_to_lds_b128 %0, %1, off"
               :: "v"(loff), "v"(gaddr) : "memory");
}

__device__ __forceinline__ void wait_async0() {
  asm volatile("s_wait_asynccnt 0x0" ::: "memory");
}

// 1 / max(sqrt(n2), 1e-12)  ==  rsqrt(n2) when n2 > 1e-24, else 1e12.
// Single v_rsq_f32 instead of v_sqrt + IEEE divide; result is re-normalized anyway.
__device__ __forceinline__ float clamped_rnorm(float n2) {
  return n2 > 1e-24f ? __builtin_amdgcn_rsqf(n2) : 1e12f;
}

__global__ __launch_bounds__(NT) void vertex_normals_fused(
    const float* __restrict__ points,   // (bs, 3, V)
    const int*   __restrict__ faces,    // (F, 3)
    const int*   __restrict__ vti,      // (V, 6)
    const float* __restrict__ vtw,      // (V, 6, 1)
    float*       __restrict__ out,      // (bs, V, 3)
    int bs, int V)
{
  __shared__ __align__(16) int   s_idx[TV * NC];     // face ids (contiguous slice of vti)
  __shared__ __align__(16) float s_w  [TV * NC];     // blend weights (contiguous slice)
  __shared__ __align__(16) int   s_fv [TV * NC * 3]; // gathered face vertex ids

  const int t  = threadIdx.x;
  const int v0 = blockIdx.x * TV;

  // ---- Phase 1: async-stage batch-invariant indices + weights into LDS ----
  // vti/vtw are row-major (V,6): this block's slice is a contiguous 768B run.
  if (v0 + TV <= V) {
    if (t < 2 * (TV * NC / 4)) {        // 96 lanes, one b128 each
      if (t < TV * NC / 4) {            // 48 x b128 -> s_idx
        async_g2l_b128((char*)s_idx + 16 * t,
                       (const char*)(vti + (size_t)v0 * NC) + 16 * t);
      } else {                          // 48 x b128 -> s_w
        const int u = t - TV * NC / 4;
        async_g2l_b128((char*)s_w + 16 * u,
                       (const char*)(vtw + (size_t)v0 * NC) + 16 * u);
      }
    }
  } else {
    // tail-safe scalar path (not taken for V % TV == 0)
    for (int e = t; e < TV * NC; e += NT) {
      int vv = v0 + e / NC; vv = vv < V ? vv : V - 1;
      const size_t off = (size_t)vv * NC + (e % NC);
      async_g2l_b32(&s_idx[e], vti + off);
      async_g2l_b32(&s_w[e],   vtw + off);
    }
  }
  wait_async0();
  __syncthreads();

  // ---- Phase 2: dependent async gather of each referenced face's 3 vertex ids ----
  if (t < TV * NC) {                    // one lane per (vertex, c)
    const int  fi   = s_idx[t];
    const int* frow = faces + (size_t)fi * 3;   // 4B-aligned -> b32 copies
    async_g2l_b32(&s_fv[t * 3 + 0], frow + 0);
    async_g2l_b32(&s_fv[t * 3 + 1], frow + 1);
    async_g2l_b32(&s_fv[t * 3 + 2], frow + 2);
  }
  wait_async0();
  __syncthreads();

  // ---- Phase 3: fused face-normal recompute + weighted blend over all batches ----
  const int vl = t % TV;       // consecutive lanes -> consecutive vertices
  const int bg = t / TV;       // batch group 0..7
  const int v  = v0 + vl;
  if (v >= V) return;

  const int    V2 = V * 2;
  const size_t V3 = (size_t)V * 3;

  for (int b = bg; b < bs; b += NT / TV) {
    const float* __restrict__ P = points + (size_t)b * V3;
    if (b + NT / TV < bs) {  // warm next batch's gather target (global_prefetch_b8)
      __builtin_prefetch(points + (size_t)(b + NT / TV) * V3 + s_fv[vl * NC * 3], 0, 0);
    }
    float ax = 0.f, ay = 0.f, az = 0.f;
#pragma unroll
    for (int c = 0; c < NC; ++c) {
      const int base = (vl * NC + c) * 3;
      const int i0 = s_fv[base], i1 = s_fv[base + 1], i2 = s_fv[base + 2];
      const float p0x = P[i0], p0y = P[V + i0], p0z = P[V2 + i0];
      const float p1x = P[i1], p1y = P[V + i1], p1z = P[V2 + i1];
      const float p2x = P[i2], p2y = P[V + i2], p2z = P[V2 + i2];
      const float e1x = p1x - p0x, e1y = p1y - p0y, e1z = p1z - p0z;
      const float e2x = p2x - p0x, e2y = p2y - p0y, e2z = p2z - p0z;
      const float cx = fmaf(e1y, e2z, -(e1z * e2y));
      const float cy = fmaf(e1z, e2x, -(e1x * e2z));
      const float cz = fmaf(e1x, e2y, -(e1y * e2x));
      const float n2 = fmaf(cx, cx, fmaf(cy, cy, cz * cz));
      const float s  = s_w[vl * NC + c] * clamped_rnorm(n2);
      ax = fmaf(cx, s, ax); ay = fmaf(cy, s, ay); az = fmaf(cz, s, az);
    }
    const float n2  = fmaf(ax, ax, fmaf(ay, ay, az * az));
    const float inv = clamped_rnorm(n2);
    const size_t o  = (size_t)b * V3 + (size_t)v * 3;
    out[o]     = ax * inv;
    out[o + 1] = ay * inv;
    out[o + 2] = az * inv;
  }
}

extern "C" void kernel_launch(void* const* d_in, const int* in_sizes, int n_in,
                              void* d_out, int out_size, void* d_ws, size_t ws_size,
                              hipStream_t stream) {
  const float* points = (const float*)d_in[0];   // (bs, 3, V) f32
  const int*   faces  = (const int*)  d_in[1];   // (F, 3) i32
  const int*   vti    = (const int*)  d_in[2];   // (V, 6) i32
  const float* vtw    = (const float*)d_in[3];   // (V, 6, 1) f32
  float*       out    = (float*)d_out;           // (bs, V, 3) f32

  const int V  = in_sizes[2] / NC;
  const int bs = (int)((long long)in_sizes[0] / ((long long)3 * V));

  const int blocks = (V + TV - 1) / TV;
  vertex_normals_fused<<<dim3(blocks), dim3(NT), 0, stream>>>(
      points, faces, vti, vtw, out, bs, V);
  (void)n_in; (void)out_size; (void)d_ws; (void)ws_size;
}